// MultiLabelGCN_63866163692191
// MI455X (gfx1250) — compile-verified
//
#include <hip/hip_runtime.h>
#include <math.h>

typedef float v2f __attribute__((ext_vector_type(2)));
typedef float v8f __attribute__((ext_vector_type(8)));

#define NJ 33
#define NE 103            // 70 directed + 33 self loops, graph 0
#define HF 256
#define NGRAPH 4096
#define NNODES (NGRAPH * NJ)
#define EPSV 1e-5f
#define FLTMAX 3.40282347e38f
#define KC 64             // K-chunk
#define SU_LD 66          // padded LDS stride for U (66 mod 64 = 2 -> low bank conflicts)

// ---------------------------------------------------------------------------
// Build normalized adjacency A[33][33] from graph 0's edge list; zero BN stats.
// ---------------------------------------------------------------------------
__global__ __launch_bounds__(128) void build_adj(const int* __restrict__ src,
                                                 const int* __restrict__ dst,
                                                 float* __restrict__ A,
                                                 float* __restrict__ s1,
                                                 float* __restrict__ s2) {
  __shared__ float deg[NJ];
  __shared__ float dinv[NJ];
  int t = threadIdx.x;
  if (t < NJ) deg[t] = 0.f;
  for (int i = t; i < NJ * NJ; i += 128) A[i] = 0.f;
  for (int i = t; i < HF; i += 128) { s1[i] = 0.f; s2[i] = 0.f; }
  __syncthreads();
  if (t < NE) atomicAdd(&deg[dst[t]], 1.0f);
  __syncthreads();
  if (t < NJ) dinv[t] = rsqrtf(deg[t]);
  __syncthreads();
  if (t < NE) {
    int i = dst[t], j = src[t];
    A[i * NJ + j] = dinv[i] * dinv[j];   // each (i,j) unique in edge list
  }
}

// ---------------------------------------------------------------------------
// Fused layer: hact = act(in) ; U = blockdiag(A) @ hact ; Y = U @ W + b
// (+ per-feature sum / sumsq accumulation for BatchNorm)
// One block = 2 graphs = 66 rows. mode: 0 = nan_to_num(x), 1 = BN+ReLU.
// ---------------------------------------------------------------------------
__global__ __launch_bounds__(256) void gcn_conv(
    const float* __restrict__ in, float* __restrict__ out,
    const float* __restrict__ W, const float* __restrict__ bias,
    const float* __restrict__ scale, const float* __restrict__ shift,
    const float* __restrict__ Aw, float* __restrict__ sum1,
    float* __restrict__ sum2, int mode, int do_stats) {
  __shared__ float sA[NJ * NJ];        //  4356 B
  __shared__ float sH[66 * KC];        // 16896 B  activation chunk
  __shared__ float sU[80 * SU_LD];     // 21120 B  aggregated chunk (padded to 80 rows)
  __shared__ float sBias[HF];
  __shared__ float sScale[HF];
  __shared__ float sShift[HF];

  const int t = threadIdx.x;
  const int lane = t & 31;
  const int wv = t >> 5;        // wave 0..7
  const int li = lane & 15;     // lane % 16
  const int hi = lane >> 4;     // lane / 16
  const int row_base = blockIdx.x * 66;

  for (int i = t; i < NJ * NJ; i += 256) sA[i] = Aw[i];
  for (int i = t; i < HF; i += 256) {
    sBias[i] = bias[i];
    if (mode == 1) { sScale[i] = scale[i]; sShift[i] = shift[i]; }
  }

  v8f acc[2][5];
  const v8f vzero = {0.f, 0.f, 0.f, 0.f, 0.f, 0.f, 0.f, 0.f};
#pragma unroll
  for (int p = 0; p < 2; p++)
#pragma unroll
    for (int m = 0; m < 5; m++) acc[p][m] = vzero;

  for (int kc = 0; kc < HF; kc += KC) {
    __syncthreads();
    // ---- load activation chunk (fused nan_to_num or BN+ReLU) ----
    for (int idx = t; idx < 66 * KC; idx += 256) {
      int r = idx >> 6;
      int c = idx & 63;
      float v = in[(row_base + r) * HF + kc + c];
      if (mode == 0) {
        if (v != v) v = 0.f;
        else if (v > FLTMAX) v = FLTMAX;
        else if (v < -FLTMAX) v = -FLTMAX;
      } else {
        v = fmaxf(v * sScale[kc + c] + sShift[kc + c], 0.f);
      }
      sH[idx] = v;
    }
    __syncthreads();
    // ---- aggregate U = A @ hact (per-graph 33x33), pad rows 66..79 with 0 ----
    {
      int f = t & 63;
      for (int i = (t >> 6); i < 80; i += 4) {
        float a = 0.f;
        if (i < 66) {
          int gl = (i >= NJ) ? 1 : 0;
          int il = i - gl * NJ;
          const float* hb = &sH[gl * NJ * KC + f];
          const float* ar = &sA[il * NJ];
#pragma unroll
          for (int j = 0; j < NJ; j++) a += ar[j] * hb[j * KC];
        }
        sU[i * SU_LD + f] = a;
      }
    }
    __syncthreads();
    // ---- WMMA GEMM: acc += U(80xKC) @ W(KCx256), wave owns 2 N-panels ----
#pragma unroll
    for (int p = 0; p < 2; p++) {
      const int ncol = ((wv + 8 * p) << 4) + li;
      for (int k0 = 0; k0 < KC; k0 += 4) {
        const int krow = kc + k0 + 2 * hi;
        v2f b;
        b.x = W[krow * HF + ncol];
        b.y = W[(krow + 1) * HF + ncol];
#pragma unroll
        for (int mt = 0; mt < 5; mt++) {
          v2f a = *(const v2f*)&sU[(mt * 16 + li) * SU_LD + k0 + 2 * hi];
          acc[p][mt] = __builtin_amdgcn_wmma_f32_16x16x4_f32(
              false, a, false, b, (short)0, acc[p][mt], false, false);
        }
      }
    }
  }

  // ---- store Y = acc + bias; accumulate BN statistics ----
#pragma unroll
  for (int p = 0; p < 2; p++) {
    const int ncol = ((wv + 8 * p) << 4) + li;
    const float bsv = sBias[ncol];
    float ps = 0.f, pq = 0.f;
#pragma unroll
    for (int mt = 0; mt < 5; mt++) {
#pragma unroll
      for (int v = 0; v < 8; v++) {
        int m = mt * 16 + v + 8 * hi;
        if (m < 66) {
          float val = acc[p][mt][v] + bsv;
          out[(row_base + m) * HF + ncol] = val;
          ps += val;
          pq += val * val;
        }
      }
    }
    if (do_stats) {
      ps += __shfl_xor(ps, 16);   // combine hi/lo half-lanes (same feature)
      pq += __shfl_xor(pq, 16);
      if (hi == 0) {
        atomicAdd(&sum1[ncol], ps);
        atomicAdd(&sum2[ncol], pq);
      }
    }
  }
}

// ---------------------------------------------------------------------------
// BN finalize: scale/shift for next layer; re-zero accumulators.
// ---------------------------------------------------------------------------
__global__ __launch_bounds__(256) void bn_finalize(
    float* __restrict__ s1, float* __restrict__ s2,
    const float* __restrict__ g, const float* __restrict__ be,
    float* __restrict__ scale, float* __restrict__ shift) {
  int t = threadIdx.x;
  const float inv = 1.0f / (float)NNODES;
  float mu = s1[t] * inv;
  float var = s2[t] * inv - mu * mu;
  float sc = g[t] * rsqrtf(var + EPSV);
  scale[t] = sc;
  shift[t] = be[t] - mu * sc;
  s1[t] = 0.f;
  s2[t] = 0.f;
}

// ---------------------------------------------------------------------------
// Global mean pool over 33 joints + final [256x4] linear head.
// ---------------------------------------------------------------------------
__global__ __launch_bounds__(256) void pool_head(
    const float* __restrict__ Y, const float* __restrict__ wc,
    const float* __restrict__ bc, float* __restrict__ outp) {
  __shared__ float sp[HF];
  int g = blockIdx.x;
  int t = threadIdx.x;
  float s = 0.f;
  const float* base = Y + (size_t)g * NJ * HF + t;
#pragma unroll
  for (int j = 0; j < NJ; j++) s += base[j * HF];
  sp[t] = s * (1.0f / (float)NJ);
  __syncthreads();
  if (t < 4) {
    float o = bc[t];
    for (int f = 0; f < HF; f++) o += sp[f] * wc[f * 4 + t];
    outp[g * 4 + t] = o;
  }
}

// ---------------------------------------------------------------------------
extern "C" void kernel_launch(void* const* d_in, const int* in_sizes, int n_in,
                              void* d_out, int out_size, void* d_ws, size_t ws_size,
                              hipStream_t stream) {
  (void)in_sizes; (void)n_in; (void)out_size; (void)ws_size;
  const float* x   = (const float*)d_in[0];
  const float* w0  = (const float*)d_in[1];
  const float* b0  = (const float*)d_in[2];
  const float* g0  = (const float*)d_in[3];
  const float* be0 = (const float*)d_in[4];
  const float* w1  = (const float*)d_in[5];
  const float* b1  = (const float*)d_in[6];
  const float* g1  = (const float*)d_in[7];
  const float* be1 = (const float*)d_in[8];
  const float* w2  = (const float*)d_in[9];
  const float* b2  = (const float*)d_in[10];
  const float* g2  = (const float*)d_in[11];
  const float* be2 = (const float*)d_in[12];
  const float* wh  = (const float*)d_in[13];
  const float* bh  = (const float*)d_in[14];
  const float* wc  = (const float*)d_in[15];
  const float* bc  = (const float*)d_in[16];
  const int*   src = (const int*)d_in[17];
  const int*   dst = (const int*)d_in[18];

  float* ws  = (float*)d_ws;
  float* A   = ws;            // 1089 (padded region 0..1151)
  float* s1  = ws + 1152;     // 256
  float* s2  = ws + 1408;     // 256
  float* sc0 = ws + 1664;
  float* sh0 = ws + 1920;
  float* sc1 = ws + 2176;
  float* sh1 = ws + 2432;
  float* sc2 = ws + 2688;
  float* sh2 = ws + 2944;
  float* bufA = ws + 4096;                         // 135168*256 floats
  float* bufB = bufA + (size_t)NNODES * HF;        // 135168*256 floats

  const dim3 cblk(256), cgrid(NGRAPH / 2);

  build_adj<<<1, 128, 0, stream>>>(src, dst, A, s1, s2);

  // layer 0: nan_to_num(x) -> conv(w0,b0), collect stats
  gcn_conv<<<cgrid, cblk, 0, stream>>>(x, bufA, w0, b0, nullptr, nullptr,
                                       A, s1, s2, /*mode=*/0, /*stats=*/1);
  bn_finalize<<<1, 256, 0, stream>>>(s1, s2, g0, be0, sc0, sh0);

  // layer 1
  gcn_conv<<<cgrid, cblk, 0, stream>>>(bufA, bufB, w1, b1, sc0, sh0,
                                       A, s1, s2, 1, 1);
  bn_finalize<<<1, 256, 0, stream>>>(s1, s2, g1, be1, sc1, sh1);

  // layer 2
  gcn_conv<<<cgrid, cblk, 0, stream>>>(bufB, bufA, w2, b2, sc1, sh1,
                                       A, s1, s2, 1, 1);
  bn_finalize<<<1, 256, 0, stream>>>(s1, s2, g2, be2, sc2, sh2);

  // head conv (wh, bh): BN2+ReLU on input, no stats
  gcn_conv<<<cgrid, cblk, 0, stream>>>(bufA, bufB, wh, bh, sc2, sh2,
                                       A, s1, s2, 1, 0);

  // mean pool + classifier
  pool_head<<<NGRAPH, 256, 0, stream>>>(bufB, wc, bc, (float*)d_out);
}